// Model_29076928593966
// MI455X (gfx1250) — compile-verified
//
#include <hip/hip_runtime.h>

// ---------------------------------------------------------------------------
// Fused SparseTSF-style kernel for MI455X (gfx1250, wave32, WMMA + TDM).
//
// rfft(12) -> per-channel Linear -> irfft(12) collapses to even/odd window
// decomposition + K=56 contraction with [w_amp | w_phase] (scaled by 0.5),
// closed-form bias (sqrt(12)*b_amp at t==0, S(t)*b_phase), and the mean
// re-centering folded into a per-(c,p) coefficient 1 - sum_s w_amp[c,p,s].
// Matrix op : V_WMMA_F32_16X16X4_F32, 14-chain per (channel, t) tile.
// Data move : TENSOR_LOAD_TO_LDS (3-D tile DMA); 6-arg (clang-23/therock) or
//             5-arg (ROCm 7.2) builtin form; float2 VMEM fallback otherwise.
// ---------------------------------------------------------------------------

#define B_    128
#define L_    336
#define C_    862
#define PRED_ 96
#define WIN_  12
#define S_IN_ 28
#define S_OUT_ 8
#define K_    56      // 2 * S_IN (even part then odd part)
#define BT_   16      // batch tile (WMMA M)
#define CT_   2       // channel tile (contiguous in memory)

typedef float v2f __attribute__((ext_vector_type(2)));
typedef float v8f __attribute__((ext_vector_type(8)));
typedef unsigned int u32x4 __attribute__((ext_vector_type(4)));
typedef int i32x4 __attribute__((ext_vector_type(4)));
typedef int i32x8 __attribute__((ext_vector_type(8)));

// ---- toolchain fingerprint probes (warnings only; never fail the build) ---
#if defined(__HIP_DEVICE_COMPILE__)
#if !__has_builtin(__builtin_amdgcn_tensor_load_to_lds)
#warning "PROBE: __builtin_amdgcn_tensor_load_to_lds MISSING on device pass"
#endif
#if !__has_builtin(__builtin_amdgcn_s_wait_tensorcnt)
#warning "PROBE: __builtin_amdgcn_s_wait_tensorcnt MISSING on device pass"
#endif
#if __has_include(<hip/amd_detail/amd_gfx1250_TDM.h>)
#warning "PROBE: <hip/amd_detail/amd_gfx1250_TDM.h> PRESENT (therock/clang-23 lane, 6-arg TDM builtin)"
#endif
#endif

// ---- TDM dispatch: 2 = 6-arg form, 1 = 5-arg form, 0 = VMEM fallback ------
#if defined(__HIP_DEVICE_COMPILE__) && __has_builtin(__builtin_amdgcn_tensor_load_to_lds)
  #if __has_include(<hip/amd_detail/amd_gfx1250_TDM.h>)
    #define TDM_MODE 2
  #else
    #define TDM_MODE 1
  #endif
#else
  #define TDM_MODE 0
#endif

// S(t) = -(2/sqrt(12)) * sum_{k=1..5} sin(pi*k*t/6)
__constant__ float c_Sc[WIN_] = {
    0.0f, -2.1547005383792515f, 0.0f, -0.5773502691896258f,
    0.0f, -0.1547005383792515f, 0.0f,  0.1547005383792515f,
    0.0f,  0.5773502691896258f, 0.0f,  2.1547005383792515f
};
#define SQRT12 3.4641016151377544f

__global__ __launch_bounds__(256) void sparse_tsf_kernel(
    const float* __restrict__ x,       // [B, L, C]
    const float* __restrict__ w_amp,   // [C, S_OUT, S_IN]
    const float* __restrict__ b_amp,   // [C, S_OUT]
    const float* __restrict__ w_phase, // [C, S_OUT, S_IN]
    const float* __restrict__ b_phase, // [C, S_OUT]
    float* __restrict__ out)           // [B, PRED, C]
{
    // TDM write order is X(=c) fastest, then Y(=b), then Z(=l): [l][b][c]
    __shared__ __align__(16) float xbuf[L_][BT_][CT_];    // 43008 B
    __shared__ __align__(16) float Wlds[CT_][16][K_];     //  7168 B  rows 8..15 = 0
    __shared__ float bamp_l[CT_][S_OUT_];
    __shared__ float bph_l [CT_][S_OUT_];
    __shared__ float wsumb [CT_][S_OUT_];                 // sum_s w_amp[c,p,s]
    __shared__ float meanb[CT_][BT_];
    __shared__ float outst[CT_][BT_][PRED_];              // 12288 B

    const int tid = (int)threadIdx.x;
    const int c0  = (int)blockIdx.x * CT_;       // channel tile base
    const int b0  = (int)blockIdx.y * BT_;       // batch tile base

    // ---- stage weights scaled by 0.5 (absorbs even/odd 1/2) --------------
    for (int f = tid; f < CT_ * S_OUT_ * S_IN_; f += 256) {
        int cc = f / (S_OUT_ * S_IN_);
        int r  = f - cc * (S_OUT_ * S_IN_);
        int p  = r / S_IN_;
        int s  = r - p * S_IN_;
        size_t g = (size_t)(c0 + cc) * (S_OUT_ * S_IN_) + (size_t)p * S_IN_ + s;
        Wlds[cc][p][s]         = 0.5f * w_amp[g];
        Wlds[cc][p][S_IN_ + s] = 0.5f * w_phase[g];
    }
    // zero-pad B rows 8..15 so all 16 N-lanes load unconditionally
    for (int f = tid; f < CT_ * 8 * K_; f += 256) {
        int cc = f / (8 * K_);
        int r  = f - cc * (8 * K_);
        Wlds[cc][8 + r / K_][r % K_] = 0.0f;
    }
    if (tid < CT_ * S_OUT_) {
        int cc = tid / S_OUT_, p = tid % S_OUT_;
        bamp_l[cc][p] = b_amp [(size_t)(c0 + cc) * S_OUT_ + p];
        bph_l [cc][p] = b_phase[(size_t)(c0 + cc) * S_OUT_ + p];
    }

    // ---- stage x[b0:b0+16, :, c0:c0+2] -----------------------------------
#if TDM_MODE
    if (tid < 32) {   // wave 0 issues one 3-D tile DMA: 2 x 16 x 336 floats
        const unsigned long long gaddr =
            (unsigned long long)(const void*)(x + ((size_t)b0 * L_) * C_ + c0);
        const unsigned int laddr =
            (unsigned int)(unsigned long long)(const void*)&xbuf[0][0][0];

        u32x4 g0;
        g0[0] = 1u;                                           // count=1 (valid user D#)
        g0[1] = laddr;                                        // lds_addr [63:32]
        g0[2] = (unsigned int)(gaddr & 0xFFFFFFFFu);          // global_addr lo
        g0[3] = (unsigned int)((gaddr >> 32) & 0x01FFFFFFu)   // global_addr hi (57b)
              | (2u << 30);                                   // type = 2 ("image")

        i32x8 g1;
        g1[0] = (int)(2u << 16);                  // data_size = 4 bytes
        g1[1] = (int)(2u << 16);                  // tensor_dim0 = 2   (c pair)
        g1[2] = (int)(16u << 16);                 // tensor_dim1 = 16  (batches)
        g1[3] = (int)(2u << 16);                  // tile_dim0   = 2
        g1[4] = (int)(16u | (336u << 16));        // tile_dim1=16, tile_dim2=336
        g1[5] = (int)(336u * 862u);               // tensor_dim0_stride = b stride
        g1[6] = (int)(862u << 16);                // tensor_dim1_stride = l stride
        g1[7] = 0;

        i32x4 g2 = {336, 0, 0, 0};                // tensor_dim2 = 336 (lines)
        i32x4 g3 = {0, 0, 0, 0};

#if TDM_MODE == 2
        i32x8 g4 = {0, 0, 0, 0, 0, 0, 0, 0};      // extra group (clang-23 form)
        __builtin_amdgcn_tensor_load_to_lds(g0, g1, g2, g3, g4, 0);
#else
        __builtin_amdgcn_tensor_load_to_lds(g0, g1, g2, g3, 0);
#endif
#if __has_builtin(__builtin_amdgcn_s_wait_tensorcnt)
        __builtin_amdgcn_s_wait_tensorcnt(0);
#else
        asm volatile("s_wait_tensorcnt 0x0" ::: "memory");
#endif
    }
#else
    for (int f = tid; f < BT_ * L_; f += 256) {
        int bb = f / L_;
        int l  = f - bb * L_;
        const float2 v = *reinterpret_cast<const float2*>(
            x + ((size_t)(b0 + bb) * L_ + l) * C_ + c0);   // C_ even, c0 even -> 8B aligned
        *reinterpret_cast<float2*>(&xbuf[l][bb][0]) = v;
    }
#endif
    __syncthreads();

    // ---- per-(b,c) means (conflict-free b64 sweeps) + w_amp row-sums -----
    if (tid < BT_) {
        float s0 = 0.0f, s1 = 0.0f;
        for (int l = 0; l < L_; ++l) {
            float2 v = *reinterpret_cast<const float2*>(&xbuf[l][tid][0]);
            s0 += v.x; s1 += v.y;
        }
        meanb[0][tid] = s0 * (1.0f / (float)L_);
        meanb[1][tid] = s1 * (1.0f / (float)L_);
    } else if (tid >= 32 && tid < 32 + CT_ * S_OUT_) {
        int idx = tid - 32;
        int cc = idx / S_OUT_, p = idx % S_OUT_;
        float s = 0.0f;
        #pragma unroll
        for (int si = 0; si < S_IN_; ++si) s += Wlds[cc][p][si];
        wsumb[cc][p] = 2.0f * s;        // undo the 0.5 staging scale
    }
    __syncthreads();

    // ---- WMMA phase: 8 waves; wave -> one channel, 3 t-tiles -------------
    const int wave = tid >> 5;
    const int lane = tid & 31;
    const int mrow = lane & 15;         // M (batch) for A, N (pred) for B/C/D
    const int hi   = lane >> 4;         // K half-select (+2) / M half-select (+8)
    const int cc   = wave >> 2;         // waves 0-3 -> cc=0, 4-7 -> cc=1

    // B fragments are task-invariant per wave: preload all 14 (ds_load_b64)
    v2f Breg[14];
    #pragma unroll
    for (int j = 0; j < 14; ++j)
        Breg[j] = *reinterpret_cast<const v2f*>(&Wlds[cc][mrow][4 * j + 2 * hi]);

    #pragma unroll
    for (int i = 0; i < 3; ++i) {
        const int t  = (wave & 3) * 3 + i;      // 0..11 within this channel
        const int t2 = (WIN_ - t) % WIN_;       // mirror sample

        // Load each raw sample pair once; steps j and j+7 share s-values.
        float ev[14], od[14];
        #pragma unroll
        for (int j = 0; j < 7; ++j) {
            int s0 = 4 * j + 2 * hi;
            float xa0 = xbuf[s0 * WIN_ + t ][mrow][cc];
            float xb0 = xbuf[s0 * WIN_ + t2][mrow][cc];
            float xa1 = xbuf[(s0 + 1) * WIN_ + t ][mrow][cc];
            float xb1 = xbuf[(s0 + 1) * WIN_ + t2][mrow][cc];
            ev[2*j]     = xa0 + xb0;  od[2*j]     = xa0 - xb0;
            ev[2*j + 1] = xa1 + xb1;  od[2*j + 1] = xa1 - xb1;
        }

        v8f acc = {};
        #pragma unroll
        for (int j = 0; j < 7; ++j) {          // K = 0..27: even part
            v2f a; a.x = ev[2*j]; a.y = ev[2*j + 1];
            acc = __builtin_amdgcn_wmma_f32_16x16x4_f32(
                false, a, false, Breg[j], (short)0, acc, false, false);
        }
        #pragma unroll
        for (int j = 0; j < 7; ++j) {          // K = 28..55: odd part
            v2f a; a.x = od[2*j]; a.y = od[2*j + 1];
            acc = __builtin_amdgcn_wmma_f32_16x16x4_f32(
                false, a, false, Breg[7 + j], (short)0, acc, false, false);
        }

        // D tile: lane column n=mrow, rows m = v + 8*hi
        if (mrow < S_OUT_) {
            const int n = mrow;
            float bias = ((t == 0) ? SQRT12 * bamp_l[cc][n] : 0.0f)
                       + c_Sc[t] * bph_l[cc][n];
            float coef = 1.0f - wsumb[cc][n];   // mean folding
            #pragma unroll
            for (int v = 0; v < 8; ++v) {
                int m = v + 8 * hi;
                outst[cc][m][n * WIN_ + t] = acc[v] + bias + meanb[cc][m] * coef;
            }
        }
    }
    __syncthreads();

    // ---- coalesced float2 stores across the channel pair -----------------
    for (int f = tid; f < BT_ * PRED_; f += 256) {
        int bb = f / PRED_;
        int pt = f - bb * PRED_;
        float2 v;
        v.x = outst[0][bb][pt];
        v.y = outst[1][bb][pt];
        *reinterpret_cast<float2*>(
            out + ((size_t)(b0 + bb) * PRED_ + pt) * C_ + c0) = v;
    }
}

extern "C" void kernel_launch(void* const* d_in, const int* in_sizes, int n_in,
                              void* d_out, int out_size, void* d_ws, size_t ws_size,
                              hipStream_t stream) {
    (void)in_sizes; (void)n_in; (void)out_size; (void)d_ws; (void)ws_size;
    const float* x       = (const float*)d_in[0];
    const float* w_amp   = (const float*)d_in[1];
    const float* b_amp   = (const float*)d_in[2];
    const float* w_phase = (const float*)d_in[3];
    const float* b_phase = (const float*)d_in[4];
    float* out = (float*)d_out;

    dim3 grid(C_ / CT_, B_ / BT_, 1);   // 431 x 8
    dim3 block(256, 1, 1);
    sparse_tsf_kernel<<<grid, block, 0, stream>>>(x, w_amp, b_amp, w_phase, b_phase, out);
}